// TemporalADDSLoss_88390426951951
// MI455X (gfx1250) — compile-verified
//
#include <hip/hip_runtime.h>
#include <hip/hip_bf16.h>

typedef __attribute__((ext_vector_type(2))) float v2f;
typedef __attribute__((ext_vector_type(8))) float v8f;

#define NB 16
#define NP 2048
#define EPSV 1e-8f

// ---------------------------------------------------------------------------
// Kernel 1: rigid transform of model points by both poses.
//  - ptsT/ptsP : (x, y, z, |p|^2) per point (A-operand + row-bias source)
//  - BpkT/BpkP : B operand pre-packed in V_WMMA_F32_16X16X4_F32 layout:
//      per 16-point column tile, 32 v2f entries:
//        [0..15]  = (-2x, -2y)      (lanes 0-15,  K=0..1)
//        [16..31] = (-2z, |p|^2)    (lanes 16-31, K=2..3)
// ---------------------------------------------------------------------------
__global__ void adds_transform(const float* __restrict__ R_t,
                               const float* __restrict__ t_t,
                               const float* __restrict__ R_p,
                               const float* __restrict__ t_p,
                               const float* __restrict__ mp,
                               float4* __restrict__ ptsT,
                               float4* __restrict__ ptsP,
                               v2f* __restrict__ BpkT,
                               v2f* __restrict__ BpkP) {
    int gid = blockIdx.x * blockDim.x + threadIdx.x;   // 0 .. NB*NP-1
    int b  = gid >> 11;                                 // / NP
    int p  = gid & (NP - 1);
    int jt = p >> 4;
    int li = p & 15;
    int pk = b * NP * 2 + jt * 32 + li;                 // packed tile base + lane

    const float* m = mp + gid * 3;
    float mx = m[0], my = m[1], mz = m[2];

    {
        const float* R = R_t + b * 9;
        const float* t = t_t + b * 3;
        float x = R[0]*mx + R[1]*my + R[2]*mz + t[0];
        float y = R[3]*mx + R[4]*my + R[5]*mz + t[1];
        float z = R[6]*mx + R[7]*my + R[8]*mz + t[2];
        float n = x*x + y*y + z*z;
        ptsT[gid] = make_float4(x, y, z, n);
        BpkT[pk]      = (v2f){-2.0f * x, -2.0f * y};
        BpkT[pk + 16] = (v2f){-2.0f * z, n};
    }
    {
        const float* R = R_p + b * 9;
        const float* t = t_p + b * 3;
        float x = R[0]*mx + R[1]*my + R[2]*mz + t[0];
        float y = R[3]*mx + R[4]*my + R[5]*mz + t[1];
        float z = R[6]*mx + R[7]*my + R[8]*mz + t[2];
        float n = x*x + y*y + z*z;
        ptsP[gid] = make_float4(x, y, z, n);
        BpkP[pk]      = (v2f){-2.0f * x, -2.0f * y};
        BpkP[pk + 16] = (v2f){-2.0f * z, n};
    }
}

// ---------------------------------------------------------------------------
// Kernel 2: per-wave 16-row strip of the squared-distance matrix via
// V_WMMA_F32_16X16X4_F32; running row-min over all 128 column tiles.
// Inner loop: 1x global_load_b64 (operand already packed) + 1x wmma + mins.
// dir=0: rows = pts_t, cols = pts_prev (min_t_to_prev)
// dir=1: rows = pts_prev, cols = pts_t (min_prev_to_t)
// ---------------------------------------------------------------------------
__global__ void adds_rowmin(const float4* __restrict__ ptsT,
                            const float4* __restrict__ ptsP,
                            const v2f* __restrict__ BpkT,
                            const v2f* __restrict__ BpkP,
                            float* __restrict__ rowmin) {
    const int wid   = (blockIdx.x * blockDim.x + threadIdx.x) >> 5;
    const int lane  = threadIdx.x & 31;
    const int li    = lane & 15;
    const int hi    = lane >> 4;     // which 16-lane half

    const int dir   = wid >> 11;     // 2048 wave-tasks per direction
    const int rem   = wid & 2047;
    const int b     = rem >> 7;      // 128 row tiles per batch
    const int itile = rem & 127;

    const float4* S1 = dir ? ptsP : ptsT;   // rows
    const v2f*    Bw = (dir ? BpkT : BpkP) + b * NP * 2;   // cols, packed

    const int base = b * NP + itile * 16;

    // A operand: row point (x,y | z,1) split across the two lane halves.
    float4 pa = S1[base + li];
    v2f a;
    a[0] = hi ? pa.z : pa.x;
    a[1] = hi ? 1.0f : pa.y;

    // C operand: |a|^2 broadcast per row (row M = v + 8*hi).
    const float* S1f = (const float*)S1;
    v8f c;
#pragma unroll
    for (int v = 0; v < 8; ++v)
        c[v] = S1f[4 * (base + v + 8 * hi) + 3];

    v8f rmin;
#pragma unroll
    for (int v = 0; v < 8; ++v) rmin[v] = 3.0e38f;

#pragma unroll 4
    for (int jt = 0; jt < NP / 16; ++jt) {
        v2f bo = Bw[jt * 32 + lane];             // pre-packed B operand

        // D = A(16x4) x B(4x16) + C  -> 16x16 tile of squared distances
        v8f d = __builtin_amdgcn_wmma_f32_16x16x4_f32(
            /*neg_a=*/false, a, /*neg_b=*/false, bo,
            /*c_mod=*/(short)0, c, /*reuse_a=*/false, /*reuse_b=*/false);

#pragma unroll
        for (int v = 0; v < 8; ++v) rmin[v] = fminf(rmin[v], d[v]);
    }

    // Min-reduce across the 16 lanes of each half (xor of bits 0..3 stays in-half).
#pragma unroll
    for (int m = 1; m <= 8; m <<= 1) {
#pragma unroll
        for (int v = 0; v < 8; ++v)
            rmin[v] = fminf(rmin[v], __shfl_xor(rmin[v], m, 32));
    }

    if (li == 0) {
        float* out = rowmin + dir * (NB * NP) + base;
#pragma unroll
        for (int v = 0; v < 8; ++v) out[v + 8 * hi] = rmin[v];
    }
}

// ---------------------------------------------------------------------------
// Kernel 3: final reduction.  loss = sum(sqrt(max(m,0)+eps)) / (B*P)
// ---------------------------------------------------------------------------
__global__ void adds_reduce(const float* __restrict__ rowmin,
                            float* __restrict__ out) {
    __shared__ float sm[256];
    const int N = 2 * NB * NP;
    float s = 0.0f;
    for (int i = threadIdx.x; i < N; i += 256)
        s += sqrtf(fmaxf(rowmin[i], 0.0f) + EPSV);
    sm[threadIdx.x] = s;
    __syncthreads();
    for (int off = 128; off > 0; off >>= 1) {
        if ((int)threadIdx.x < off) sm[threadIdx.x] += sm[threadIdx.x + off];
        __syncthreads();
    }
    if (threadIdx.x == 0) out[0] = sm[0] * (1.0f / (float)(NB * NP));
}

// ---------------------------------------------------------------------------
extern "C" void kernel_launch(void* const* d_in, const int* in_sizes, int n_in,
                              void* d_out, int out_size, void* d_ws, size_t ws_size,
                              hipStream_t stream) {
    const float* R_t = (const float*)d_in[0];
    const float* t_t = (const float*)d_in[1];
    const float* R_p = (const float*)d_in[2];
    const float* t_p = (const float*)d_in[3];
    const float* mp  = (const float*)d_in[4];

    char* ws = (char*)d_ws;
    size_t off = 0;
    float4* ptsT   = (float4*)(ws + off); off += (size_t)NB * NP * sizeof(float4);
    float4* ptsP   = (float4*)(ws + off); off += (size_t)NB * NP * sizeof(float4);
    v2f*    BpkT   = (v2f*)   (ws + off); off += (size_t)NB * NP * 2 * sizeof(v2f);
    v2f*    BpkP   = (v2f*)   (ws + off); off += (size_t)NB * NP * 2 * sizeof(v2f);
    float*  rowmin = (float*) (ws + off);

    // 1) transform + operand packing: one thread per point
    adds_transform<<<(NB * NP) / 256, 256, 0, stream>>>(R_t, t_t, R_p, t_p, mp,
                                                        ptsT, ptsP, BpkT, BpkP);

    // 2) row-min strips: 2 dirs * 16 batches * 128 tiles = 4096 waves,
    //    8 waves (256 threads) per block -> 512 blocks
    adds_rowmin<<<512, 256, 0, stream>>>(ptsT, ptsP, BpkT, BpkP, rowmin);

    // 3) scalar reduction
    adds_reduce<<<1, 256, 0, stream>>>(rowmin, (float*)d_out);
}